// SlotCovarianceBlock_8589934592088
// MI455X (gfx1250) — compile-verified
//
#include <hip/hip_runtime.h>
#include <hip/hip_bf16.h>

typedef __attribute__((ext_vector_type(2))) float v2f;
typedef __attribute__((ext_vector_type(8))) float v8f;

#define B_Q     512
#define K_SLOTS 16
#define C_DIM   256
#define NCLS    64
#define SHOTS   5
#define NSUP    (SHOTS * K_SLOTS)   // 80 support rows per class
#define QROWS   (B_Q * K_SLOTS)     // 8192 query rows

__device__ __forceinline__ float wave_reduce_sum(float x) {
#pragma unroll
  for (int off = 16; off > 0; off >>= 1)
    x += __shfl_xor(x, off, 32);
  return x;
}

__device__ __forceinline__ v8f vzero8() { v8f z = {}; return z; }

// ---------------------------------------------------------------------------
// Kernel 0: L2-normalize query rows (8192 rows x 256). One wave per row.
// ---------------------------------------------------------------------------
__global__ void __launch_bounds__(256)
slotcov_normalize_q(const float* __restrict__ q_in, float* __restrict__ q_out) {
  const int wave = threadIdx.x >> 5;
  const int lane = threadIdx.x & 31;
  const int row  = blockIdx.x * 8 + wave;           // 1024 blocks * 8 waves
  const float* src = q_in  + (size_t)row * C_DIM;
  float*       dst = q_out + (size_t)row * C_DIM;
  float v[8];
  float ss = 0.f;
#pragma unroll
  for (int j = 0; j < 8; ++j) { v[j] = src[lane + 32 * j]; ss += v[j] * v[j]; }
  ss = wave_reduce_sum(ss);
  const float sc = 1.0f / fmaxf(sqrtf(ss), 1e-12f);
#pragma unroll
  for (int j = 0; j < 8; ++j) dst[lane + 32 * j] = v[j] * sc;
}

// ---------------------------------------------------------------------------
// Kernel 1: per class -> normalize 80 support rows, mean-center in LDS, then
// cov = centered^T @ centered / 79 + 1e-3*I via V_WMMA_F32_16X16X4_F32.
// One block (8 waves, 256 thr) per class. LDS: 80*256*4 = 80 KB.
// ---------------------------------------------------------------------------
__global__ void __launch_bounds__(256)
slotcov_class_cov(const float* __restrict__ sup, float* __restrict__ cov) {
  __shared__ float S[NSUP * C_DIM];
  const int n    = blockIdx.x;
  const int wave = threadIdx.x >> 5;
  const int lane = threadIdx.x & 31;
  const float* base = sup + (size_t)n * NSUP * C_DIM;

  // 1) row L2-normalization: wave w handles rows [w*10, w*10+10)
  for (int i = 0; i < 10; ++i) {
    const int r = wave * 10 + i;
    const float* src = base + r * C_DIM;
    float v[8];
    float ss = 0.f;
#pragma unroll
    for (int j = 0; j < 8; ++j) { v[j] = src[lane + 32 * j]; ss += v[j] * v[j]; }
    ss = wave_reduce_sum(ss);
    const float sc = 1.0f / fmaxf(sqrtf(ss), 1e-12f);
#pragma unroll
    for (int j = 0; j < 8; ++j) S[r * C_DIM + lane + 32 * j] = v[j] * sc;
  }
  __syncthreads();

  // 2) mean-center: thread t owns column t
  {
    const int c = threadIdx.x;
    float m = 0.f;
    for (int r = 0; r < NSUP; ++r) m += S[r * C_DIM + c];
    m *= (1.0f / NSUP);
    for (int r = 0; r < NSUP; ++r) S[r * C_DIM + c] -= m;
  }
  __syncthreads();

  // 3) covariance tiles with f32 WMMA. 256 tiles of 16x16; wave w does
  //    tiles {w, w+8, ...}. K dim = 80 = 20 WMMA steps of 4.
  const float inv_n = 1.0f / ((float)(NSUP - 1) + 1e-8f);
  const int m16   = lane & 15;
  const int koff  = (lane >> 4) * 2;        // lanes 0-15 -> K 0,1 ; 16-31 -> K 2,3
  const int rowhi = (lane >> 4) << 3;       // C/D layout: +8 row offset for hi lanes
  float* out = cov + (size_t)n * C_DIM * C_DIM;

  for (int t = 0; t < 32; ++t) {
    const int tile = wave + 8 * t;
    const int ci = (tile >> 4) * 16;        // output row block (channel i)
    const int cj = (tile & 15) * 16;        // output col block (channel j)
    v8f acc = vzero8();
#pragma unroll
    for (int k0 = 0; k0 < NSUP; k0 += 4) {
      v2f a, b;
      a.x = S[(k0 + koff)     * C_DIM + ci + m16];   // A = centered^T (M=ci+m, K)
      a.y = S[(k0 + koff + 1) * C_DIM + ci + m16];
      b.x = S[(k0 + koff)     * C_DIM + cj + m16];   // B = centered   (K, N=cj+n)
      b.y = S[(k0 + koff + 1) * C_DIM + cj + m16];
      acc = __builtin_amdgcn_wmma_f32_16x16x4_f32(false, a, false, b,
                                                  (short)0, acc, false, false);
    }
#pragma unroll
    for (int r = 0; r < 8; ++r) {
      const int row = ci + r + rowhi;
      const int col = cj + m16;
      float vv = acc[r] * inv_n;
      if (row == col) vv += 1e-3f;          // ridge
      out[row * C_DIM + col] = vv;
    }
  }
}

// ---------------------------------------------------------------------------
// Kernel 2: in-place Gauss-Jordan inverse of each 256x256 SPD matrix.
// One block per class; pivot row cached in LDS; wave-per-row elimination.
// Matrices are L2-resident (16 MB total << 192 MB L2).
// ---------------------------------------------------------------------------
__global__ void __launch_bounds__(256)
slotcov_gj_inverse(float* __restrict__ A_all) {
  __shared__ float prow[C_DIM];
  float* A = A_all + (size_t)blockIdx.x * C_DIM * C_DIM;
  const int t    = threadIdx.x;
  const int wave = t >> 5;
  const int lane = t & 31;

  for (int k = 0; k < C_DIM; ++k) {
    // ---- phase 1: scale pivot row (read everything before anyone writes)
    const float piv = A[k * C_DIM + k];
    const float v   = A[k * C_DIM + t];
    __syncthreads();
    const float pinv = 1.0f / piv;
    const float nv   = (t == k) ? pinv : v * pinv;
    prow[t] = nv;
    A[k * C_DIM + t] = nv;
    __syncthreads();
    // ---- phase 2: eliminate column k from all other rows; wave w owns
    //      rows {w, w+8, w+16, ...} so each row is touched by one wave only.
    const float pk = prow[k];                // == 1/piv
    for (int ii = 0; ii < 32; ++ii) {
      const int i = wave + 8 * ii;
      if (i == k) continue;
      const float f = A[i * C_DIM + k];      // read-before-write within wave
#pragma unroll
      for (int j8 = 0; j8 < 8; ++j8) {
        const int col = j8 * 32 + lane;
        const float cur = A[i * C_DIM + col];
        const float res = (col == k) ? (-f * pk) : fmaf(-f, prow[col], cur);
        A[i * C_DIM + col] = res;
      }
    }
    __syncthreads();
  }
}

// ---------------------------------------------------------------------------
// Kernel 3: out[b][n] = -sum_{k,d} (Q_b @ invCov_n)[k,d] * Q_b[k,d]
// Grid (64 b-groups, 64 classes); block = 8 waves; wave w owns b = bx*8+w.
// The 16-row A tile (rows b*16..b*16+15) is exactly the 16 slots of one b,
// so each wave produces one scalar. 16 v8f accumulators span d=0..255;
// invCov c-slabs (64x256 = 64 KB) staged in LDS, shared by all 8 waves.
// 1024 f32 WMMAs per wave -> ~69 GFLOP total across the grid.
// ---------------------------------------------------------------------------
__global__ void __launch_bounds__(256)
slotcov_scores(const float* __restrict__ Q, const float* __restrict__ Inv,
               float* __restrict__ out) {
  __shared__ float Bs[64 * C_DIM];          // 64 KB c-slab of invCov_n
  const int n    = blockIdx.y;
  const int wave = threadIdx.x >> 5;
  const int lane = threadIdx.x & 31;
  const int b    = blockIdx.x * 8 + wave;
  const float* qrow0 = Q + (size_t)(b * K_SLOTS) * C_DIM;   // 16x256 A panel
  const float* M     = Inv + (size_t)n * C_DIM * C_DIM;

  const int m16   = lane & 15;
  const int koff  = (lane >> 4) * 2;
  const int rowhi = (lane >> 4) << 3;

  v8f acc[16];
#pragma unroll
  for (int dt = 0; dt < 16; ++dt) acc[dt] = vzero8();

  for (int cs = 0; cs < 4; ++cs) {
    __syncthreads();
    {   // cooperative stage of invCov rows [cs*64, cs*64+64) into LDS
      const float4* src = (const float4*)(M + (size_t)cs * 64 * C_DIM);
      float4*       dst = (float4*)Bs;
      for (int idx = threadIdx.x; idx < (64 * C_DIM) / 4; idx += 256)
        dst[idx] = src[idx];
    }
    __syncthreads();

    for (int c4 = 0; c4 < 16; ++c4) {
      const int c0 = c4 * 4;                 // c offset within slab
      const int cg = cs * 64 + c0;           // global c offset
      v2f a;                                 // A: M = slot m16, K = cg+koff(+1)
      const float* ap = qrow0 + m16 * C_DIM + cg + koff;
      a.x = ap[0];
      a.y = ap[1];
#pragma unroll
      for (int dt = 0; dt < 16; ++dt) {      // B: K rows from LDS, N = d tile
        v2f bb;
        bb.x = Bs[(c0 + koff)     * C_DIM + dt * 16 + m16];
        bb.y = Bs[(c0 + koff + 1) * C_DIM + dt * 16 + m16];
        acc[dt] = __builtin_amdgcn_wmma_f32_16x16x4_f32(false, a, false, bb,
                                                        (short)0, acc[dt],
                                                        false, false);
      }
    }
  }

  // T (x) Q contraction over the whole 16x256 tile, then wave32 reduce.
  float partial = 0.f;
#pragma unroll
  for (int dt = 0; dt < 16; ++dt) {
#pragma unroll
    for (int r = 0; r < 8; ++r) {
      const float qv = qrow0[(r + rowhi) * C_DIM + dt * 16 + m16];
      partial = fmaf(acc[dt][r], qv, partial);
    }
  }
  const float s = wave_reduce_sum(partial);
  if (lane == 0) out[b * NCLS + n] = -s;
}

// ---------------------------------------------------------------------------
extern "C" void kernel_launch(void* const* d_in, const int* in_sizes, int n_in,
                              void* d_out, int out_size, void* d_ws, size_t ws_size,
                              hipStream_t stream) {
  (void)in_sizes; (void)n_in; (void)out_size; (void)ws_size;
  const float* query   = (const float*)d_in[0];   // (512,16,256) f32
  const float* support = (const float*)d_in[1];   // (64,5,16,256) f32
  float* out = (float*)d_out;                     // (512,64) f32

  // ws layout: [ Qnorm : 8192*256 f32 | COV/INV : 64*256*256 f32 ]  ~25 MB
  float* wsf  = (float*)d_ws;
  float* Qn   = wsf;
  float* COV  = wsf + (size_t)QROWS * C_DIM;

  slotcov_normalize_q<<<QROWS / 8, 256, 0, stream>>>(query, Qn);
  slotcov_class_cov  <<<NCLS,      256, 0, stream>>>(support, COV);
  slotcov_gj_inverse <<<NCLS,      256, 0, stream>>>(COV);
  slotcov_scores     <<<dim3(B_Q / 8, NCLS), 256, 0, stream>>>(Qn, COV, out);
}